// HierarchicalTargetAttention_38062000177793
// MI455X (gfx1250) — compile-verified
//
#include <hip/hip_runtime.h>

typedef __attribute__((ext_vector_type(2))) float v2f;
typedef __attribute__((ext_vector_type(4))) float v4f;
typedef __attribute__((ext_vector_type(8))) float v8f;

namespace {
constexpr int kB  = 4;
constexpr int kD  = 512;
constexpr int kL  = 4096;
constexpr int kN1 = 1167;
constexpr int kN2 = 8921;

constexpr int SS = kL + 4;   // score-strip LDS row stride (16B-aligned rows, conflict-free)
constexpr int QS = kD + 4;   // Q-block LDS row stride     (16B-aligned rows, conflict-free)
}

__device__ __forceinline__ v8f wmma_f32(v2f a, v2f b, v8f c) {
  // D = A(16x4,f32) * B(4x16,f32) + C(16x16,f32)  -> v_wmma_f32_16x16x4_f32
  return __builtin_amdgcn_wmma_f32_16x16x4_f32(false, a, false, b, (short)0, c, false, false);
}

// ---------------------------------------------------------------------------
// Kernel 1: fused scores + softmax.
//   A_out[b, n, l] = softmax_l( sum_d Q[b?, n, d] * H[b, d, l] )
// One workgroup = one 16-row strip of N. The whole 16 x 4096 score strip lives
// in LDS (256 KB of the 320 KB WGP LDS): E -> softmax -> A in a single pass.
// Each wave drives TWO independent WMMA accumulator chains (adjacent L tiles)
// sharing one A operand, for ILP at the low occupancy LDS usage forces.
// STREAM=true marks the score output (and Q reads) non-temporal so the A2
// stream does not evict the L2-resident H (33 MB must stay hot in 192 MB L2).
// ---------------------------------------------------------------------------
template <bool STREAM>
__global__ __launch_bounds__(256)
void hta_attn_scores(const float* __restrict__ Qbase, long qBatchStride,
                     const float* __restrict__ Hbase,
                     float* __restrict__ Abase, int nRows)
{
  extern __shared__ float smem[];
  float* scoresS = smem;                 // [16][SS]
  float* Qs      = smem + 16 * SS;       // [16][QS]
  float* red     = Qs + 16 * QS;         // [16][16] partial reductions
  float* rowstat = red + 256;            // [0..15]=row max, [16..31]=1/sum

  const int tid  = threadIdx.x;
  const int b    = blockIdx.y;
  const int row0 = blockIdx.x * 16;

  const float* Q = Qbase + (long)b * qBatchStride + (long)row0 * kD;
  const float* H = Hbase + (long)b * kD * (long)kL;
  float*       A = Abase + ((long)b * nRows + row0) * (long)kL;

  // Stage the 16 x 512 query block as float4 (zero-pad rows past nRows).
  for (int idx4 = tid; idx4 < 16 * (kD / 4); idx4 += 256) {
    int r  = idx4 >> 7;          // /128 float4 per row
    int c4 = idx4 & 127;
    v4f v = {0.0f, 0.0f, 0.0f, 0.0f};
    if (row0 + r < nRows) {
      const v4f* src = reinterpret_cast<const v4f*>(Q + (long)r * kD + c4 * 4);
      if constexpr (STREAM) v = __builtin_nontemporal_load(src);
      else                  v = *src;
    }
    *reinterpret_cast<v4f*>(&Qs[r * QS + c4 * 4]) = v;
  }
  __syncthreads();

  const int wave = tid >> 5;
  const int lane = tid & 31;
  const int la   = lane & 15;   // M-row (A) / N-col (B,C) index
  const int hi   = lane >> 4;   // half-wave select
  const int koff = hi * 2;      // K offset for A/B operands in upper half-wave

  // Each wave owns strided PAIRS of 16-wide L tiles (2 independent chains).
  for (int ltp = wave * 2; ltp < kL / 16; ltp += 16) {
    const int l0 = ltp * 16;
    v8f acc0 = {0, 0, 0, 0, 0, 0, 0, 0};
    v8f acc1 = {0, 0, 0, 0, 0, 0, 0, 0};
    const float* Hp0 = H + l0 + la;
    const float* Hp1 = Hp0 + 16;
#pragma unroll 4
    for (int k0 = 0; k0 < kD; k0 += 4) {
      v2f a, b0, b1;
      a.x  = Qs[la * QS + k0 + koff];          // A[m, k0+koff]
      a.y  = Qs[la * QS + k0 + koff + 1];      // A[m, k0+koff+1]
      b0.x = Hp0[(long)(k0 + koff) * kL];      // B tile0
      b0.y = Hp0[(long)(k0 + koff + 1) * kL];
      b1.x = Hp1[(long)(k0 + koff) * kL];      // B tile1
      b1.y = Hp1[(long)(k0 + koff + 1) * kL];
      acc0 = wmma_f32(a, b0, acc0);
      acc1 = wmma_f32(a, b1, acc1);
    }
#pragma unroll
    for (int r = 0; r < 8; ++r) {              // C row = r + 8*hi, col = la
      scoresS[(r + 8 * hi) * SS + l0 + la]      = acc0[r];
      scoresS[(r + 8 * hi) * SS + l0 + 16 + la] = acc1[r];
    }
  }
  __syncthreads();

  // ---- softmax over L (16 threads per row) ----
  const int r = tid >> 4;       // row 0..15
  const int j = tid & 15;       // lane-in-row
  float mx = -3.0e38f;
  for (int c = j; c < kL; c += 16) mx = fmaxf(mx, scoresS[r * SS + c]);
  red[r * 16 + j] = mx;
  __syncthreads();
  if (tid < 16) {
    float v = red[tid * 16];
#pragma unroll
    for (int t = 1; t < 16; ++t) v = fmaxf(v, red[tid * 16 + t]);
    rowstat[tid] = v;
  }
  __syncthreads();
  const float rmax = rowstat[r];
  float s = 0.0f;
  for (int c = j; c < kL; c += 16) {
    float e = __expf(scoresS[r * SS + c] - rmax);
    scoresS[r * SS + c] = e;
    s += e;
  }
  red[r * 16 + j] = s;
  __syncthreads();
  if (tid < 16) {
    float v = 0.0f;
#pragma unroll
    for (int t = 0; t < 16; ++t) v += red[tid * 16 + t];
    rowstat[16 + tid] = 1.0f / v;
  }
  __syncthreads();

  // Coalesced normalized write-out, 16B per lane (this is the HBM-bound path).
  for (int idx4 = tid; idx4 < 16 * (kL / 4); idx4 += 256) {
    int rr = idx4 >> 10;         // /1024 float4 per row
    int c4 = idx4 & 1023;
    if (row0 + rr < nRows) {
      v4f v = *reinterpret_cast<const v4f*>(&scoresS[rr * SS + c4 * 4]);
      float sc = rowstat[16 + rr];
      v.x *= sc; v.y *= sc; v.z *= sc; v.w *= sc;
      v4f* dst = reinterpret_cast<v4f*>(A + (long)rr * kL + c4 * 4);
      if constexpr (STREAM) __builtin_nontemporal_store(v, dst);
      else                  *dst = v;
    }
  }
}

// ---------------------------------------------------------------------------
// Kernel 2: context GEMM  C[b, m, d] = sum_l P[b, m, l] * H[b, d, l]
// (i.e. C = P * H^T). 64x64 C macro-tile, K-chunk 64.
// H tile is stored PAIR-INTERLEAVED along K: element (l,d) lives at
// Hts[l>>1][d*2 + (l&1)], so each WMMA B operand {H[l],H[l+1]} is one aligned
// ds_load_b64 straight into an even VGPR pair (no repack moves).
// STREAM=true: P (the 585 MB A2 stream, read once) is loaded non-temporal and
// C (final output) stored non-temporal, protecting H's L2 residency.
// ---------------------------------------------------------------------------
template <bool STREAM>
__global__ __launch_bounds__(256)
void hta_ctx_gemm(const float* __restrict__ Pbase,  // [B, M, L]
                  const float* __restrict__ Hbase,  // [B, D, L]
                  float* __restrict__ Cbase,        // [B, M, D]
                  int M)
{
  __shared__ float Ps[64][68];     // [m][l]   (rows 16B aligned)
  __shared__ float Hts[32][130];   // (l,d) -> [l>>1][d*2 + (l&1)], pairs 8B aligned

  const int tid = threadIdx.x;
  const int b   = blockIdx.z;
  const int m0  = blockIdx.x * 64;
  const int d0  = blockIdx.y * 64;

  const float* P = Pbase + (long)b * M * (long)kL;
  const float* H = Hbase + (long)b * kD * (long)kL;
  float*       C = Cbase + (long)b * M * (long)kD;

  const int wave = tid >> 5;
  const int lane = tid & 31;
  const int la   = lane & 15;
  const int hi   = lane >> 4;
  const int koff = hi * 2;
  const int mi   = (wave & 3) * 16;      // M subtile
  const int ni   = (wave >> 2) * 32;     // pair of N subtiles

  v8f acc0 = {0, 0, 0, 0, 0, 0, 0, 0};
  v8f acc1 = {0, 0, 0, 0, 0, 0, 0, 0};

  for (int l0 = 0; l0 < kL; l0 += 64) {
    __syncthreads();
    // P tile: 64x64 = 1024 float4, b128 in / b128 to LDS.
    for (int idx4 = tid; idx4 < 64 * 16; idx4 += 256) {
      int rr = idx4 >> 4;
      int c4 = idx4 & 15;
      v4f v = {0.0f, 0.0f, 0.0f, 0.0f};
      if (m0 + rr < M) {
        const v4f* src = reinterpret_cast<const v4f*>(P + (long)(m0 + rr) * kL + l0 + c4 * 4);
        if constexpr (STREAM) v = __builtin_nontemporal_load(src);
        else                  v = *src;
      }
      *reinterpret_cast<v4f*>(&Ps[rr][c4 * 4]) = v;
    }
    // H tile: coalesced global read (RT: keep L2-resident), pair-interleaved store.
    for (int idx = tid; idx < 64 * 64; idx += 256) {
      int rr = idx >> 6;        // d-local
      int cc = idx & 63;        // l-local
      Hts[cc >> 1][rr * 2 + (cc & 1)] = H[(long)(d0 + rr) * kL + l0 + cc];
    }
    __syncthreads();
#pragma unroll
    for (int k = 0; k < 64; k += 4) {
      v2f a  = *reinterpret_cast<const v2f*>(&Ps[mi + la][k + koff]);            // 8B aligned
      v2f b0 = *reinterpret_cast<const v2f*>(&Hts[(k + koff) >> 1][(ni + la) * 2]);
      v2f b1 = *reinterpret_cast<const v2f*>(&Hts[(k + koff) >> 1][(ni + 16 + la) * 2]);
      acc0 = wmma_f32(a, b0, acc0);
      acc1 = wmma_f32(a, b1, acc1);
    }
  }

#pragma unroll
  for (int r = 0; r < 8; ++r) {
    int grow = m0 + mi + r + 8 * hi;
    if (grow < M) {
      float* p0 = &C[(long)grow * kD + d0 + ni + la];
      float* p1 = &C[(long)grow * kD + d0 + ni + 16 + la];
      if constexpr (STREAM) {
        __builtin_nontemporal_store(acc0[r], p0);
        __builtin_nontemporal_store(acc1[r], p1);
      } else {
        *p0 = acc0[r];
        *p1 = acc1[r];
      }
    }
  }
}

// ---------------------------------------------------------------------------
// Kernel 3: projection GEMM + modulation.
//   O[b, m, d] = Scale[m, d] * sum_n A12[m, n] * Bm[b, n, d]
// B tile pair-interleaved along K like hta_ctx_gemm; A tile scalar-staged
// (K = N1 = 1167 ragged edge). Scale (73 MB, read once) loaded non-temporal.
// ---------------------------------------------------------------------------
__global__ __launch_bounds__(256)
void hta_mod_gemm(const float* __restrict__ A12,    // [M, K]   (Q12_w)
                  const float* __restrict__ Bbase,  // [B, K, D] (C1')
                  const float* __restrict__ Scale,  // [M, D]   (Q2_w)
                  float* __restrict__ Obase,        // [B, M, D]
                  int M, int K)
{
  __shared__ float As[64][65];     // [m][k]
  __shared__ float Bs[32][130];    // (k,d) -> [k>>1][d*2 + (k&1)]

  const int tid = threadIdx.x;
  const int b   = blockIdx.z;
  const int m0  = blockIdx.x * 64;
  const int d0  = blockIdx.y * 64;

  const float* Bm = Bbase + (long)b * K * (long)kD;
  float*       O  = Obase + (long)b * M * (long)kD;

  const int wave = tid >> 5;
  const int lane = tid & 31;
  const int la   = lane & 15;
  const int hi   = lane >> 4;
  const int koff = hi * 2;
  const int mi   = (wave & 3) * 16;
  const int ni   = (wave >> 2) * 32;

  v8f acc0 = {0, 0, 0, 0, 0, 0, 0, 0};
  v8f acc1 = {0, 0, 0, 0, 0, 0, 0, 0};

  for (int k0 = 0; k0 < K; k0 += 64) {
    __syncthreads();
    for (int idx = tid; idx < 64 * 64; idx += 256) {
      int rr = idx >> 6;
      int cc = idx & 63;
      As[rr][cc] = (m0 + rr < M && k0 + cc < K) ? A12[(long)(m0 + rr) * K + k0 + cc] : 0.0f;
      Bs[rr >> 1][cc * 2 + (rr & 1)] =
          (k0 + rr < K) ? Bm[(long)(k0 + rr) * kD + d0 + cc] : 0.0f;
    }
    __syncthreads();
#pragma unroll
    for (int k = 0; k < 64; k += 4) {
      v2f a;
      a.x = As[mi + la][k + koff];
      a.y = As[mi + la][k + koff + 1];
      v2f b0 = *reinterpret_cast<const v2f*>(&Bs[(k + koff) >> 1][(ni + la) * 2]);
      v2f b1 = *reinterpret_cast<const v2f*>(&Bs[(k + koff) >> 1][(ni + 16 + la) * 2]);
      acc0 = wmma_f32(a, b0, acc0);
      acc1 = wmma_f32(a, b1, acc1);
    }
  }

#pragma unroll
  for (int r = 0; r < 8; ++r) {
    int grow = m0 + mi + r + 8 * hi;
    if (grow < M) {
      int c0 = d0 + ni + la;
      int c1 = d0 + ni + 16 + la;
      float s0 = __builtin_nontemporal_load(&Scale[(long)grow * kD + c0]);
      float s1 = __builtin_nontemporal_load(&Scale[(long)grow * kD + c1]);
      O[(long)grow * kD + c0] = acc0[r] * s0;
      O[(long)grow * kD + c1] = acc1[r] * s1;
    }
  }
}

// ---------------------------------------------------------------------------
// Host launcher. d_out = [C2 (B*N2*D) | A2 (B*N2*L)].
// Scratch lives inside d_out regions that are later overwritten:
//   A1 scratch  -> head of A2 region (consumed by stage 1b)
//   C1' scratch -> A2 region after A1 (consumed by stage 2)
//   Q2 scratch  -> C2 region (consumed by stage 3a, overwritten by 3b)
// Cache policy: level-1 intermediates (A1: 76 MB) are kept RT so they stay in
// the 192 MB L2 next to H (33 MB); the level-2 A2 stream (2x 585 MB) is
// non-temporal end-to-end so H never leaves L2.
// ---------------------------------------------------------------------------
extern "C" void kernel_launch(void* const* d_in, const int* in_sizes, int n_in,
                              void* d_out, int out_size, void* d_ws, size_t ws_size,
                              hipStream_t stream) {
  (void)in_sizes; (void)n_in; (void)out_size; (void)d_ws; (void)ws_size;

  const float* H   = (const float*)d_in[0];   // [B, D, L]
  const float* Q1w = (const float*)d_in[1];   // [N1, D]
  const float* Q2w = (const float*)d_in[2];   // [N2, D]
  const float* Q12 = (const float*)d_in[3];   // [N2, N1]

  float* out   = (float*)d_out;
  float* C2    = out;                                   // B*N2*D
  float* A2    = out + (size_t)kB * kN2 * kD;           // B*N2*L
  float* A1scr = A2;                                    // B*N1*L  (scratch)
  float* C1p   = A2 + (size_t)kB * kN1 * kL;            // B*N1*D  (scratch)
  float* Q2scr = C2;                                    // B*N2*D  (scratch)

  const size_t attnSmem = (size_t)(16 * SS + 16 * QS + 256 + 32) * sizeof(float); // ~296 KB
  hipFuncSetAttribute(reinterpret_cast<const void*>(hta_attn_scores<false>),
                      hipFuncAttributeMaxDynamicSharedMemorySize, (int)attnSmem);
  hipFuncSetAttribute(reinterpret_cast<const void*>(hta_attn_scores<true>),
                      hipFuncAttributeMaxDynamicSharedMemorySize, (int)attnSmem);

  dim3 blk(256, 1, 1);

  // 1a: A1 = softmax(Q1_w . H)          (RT: A1 + H fit in L2 together)
  hta_attn_scores<false><<<dim3((kN1 + 15) / 16, kB), blk, attnSmem, stream>>>(
      Q1w, 0L, H, A1scr, kN1);
  // 1b: C1 = A1 . H^T                   (RT: A1 read from L2)
  hta_ctx_gemm<false><<<dim3((kN1 + 63) / 64, kD / 64, kB), blk, 0, stream>>>(
      A1scr, H, C1p, kN1);
  // 2:  Q2 = Q2_w * (Q12 . C1)
  hta_mod_gemm<<<dim3((kN2 + 63) / 64, kD / 64, kB), blk, 0, stream>>>(
      Q12, C1p, Q2w, Q2scr, kN2, kN1);
  // 3a: A2 = softmax(Q2 . H)            (NT stream out: protect H in L2)
  hta_attn_scores<true><<<dim3((kN2 + 15) / 16, kB), blk, attnSmem, stream>>>(
      Q2scr, (long)kN2 * kD, H, A2, kN2);
  // 3b: C2 = A2 . H^T                   (NT stream in/out: protect H in L2)
  hta_ctx_gemm<true><<<dim3((kN2 + 63) / 64, kD / 64, kB), blk, 0, stream>>>(
      A2, H, C2, kN2);
}